// SelfAttention_56341380989371
// MI455X (gfx1250) — compile-verified
//
#include <hip/hip_runtime.h>

typedef __attribute__((ext_vector_type(2))) float v2f;
typedef __attribute__((ext_vector_type(8))) float v8f;

namespace {
constexpr int B_ = 2, S_ = 1024, E_ = 1024, H_ = 16, D_ = 64;
constexpr int QKV_LD = 3 * E_;       // 3072
constexpr float SCALE_ = 0.125f;     // 1/sqrt(64)
constexpr float NEGM_ = -10000.0f;
}

// D = A(16x4 f32) * B(4x16 f32) + C(16x16 f32), full f32 precision.
__device__ __forceinline__ v8f wmma4(v2f a, v2f b, v8f c) {
  return __builtin_amdgcn_wmma_f32_16x16x4_f32(
      /*neg_a=*/false, a, /*neg_b=*/false, b,
      /*c_mod=*/(short)0, c, /*reuse_a=*/false, /*reuse_b=*/false);
}

// Async global->LDS copies (CDNA5, ASYNCcnt-tracked).
__device__ __forceinline__ void async_ld_b32(unsigned lds_byte_off,
                                             const float* gptr) {
  asm volatile("global_load_async_to_lds_b32 %0, %1, off"
               :: "v"(lds_byte_off), "v"(gptr) : "memory");
}
__device__ __forceinline__ void async_ld_b128(unsigned lds_byte_off,
                                              const float* gptr) {
  asm volatile("global_load_async_to_lds_b128 %0, %1, off"
               :: "v"(lds_byte_off), "v"(gptr) : "memory");
}
#define S_WAIT_ASYNC(imm) asm volatile("s_wait_asynccnt " #imm ::: "memory")

// LDS offset of a __shared__ object = low 32 bits of its generic address
// (ISA: LDS aperture maps addr[31:0] -> LDS offset).
__device__ __forceinline__ unsigned lds_off(const void* p) {
  return (unsigned)(unsigned long long)p;
}

// 8B-aligned contiguous pair load (lowers to ds_load_b64 -> direct WMMA src).
__device__ __forceinline__ v2f ld2(const float* p) {
  return *reinterpret_cast<const v2f*>(p);
}

// ---------------------------------------------------------------------------
// Generic C[M,N] = A[M,K] @ B[K,N] + bias.  Block tile 128x128, 256 thr
// (8 waves arranged 4x2; each wave computes 32x64 = 2x4 WMMA tiles).
// Double-buffered LDS staged via async global->LDS copies.
// A tile: row stride 20 (b64-aligned fragments, conflict-free columns).
// B tile: K-row pairs interleaved per column (one b64 per fragment).
// Per stage per thread: 2 async b128 (A) + 8 async b32 (B) = 10 outstanding.
// ---------------------------------------------------------------------------
__global__ __launch_bounds__(256, 2) void gemm_bias_128x128(
    const float* __restrict__ A, const float* __restrict__ Bm,
    const float* __restrict__ bias, float* __restrict__ C,
    int M, int N, int K) {
  constexpr int ALD  = 20;            // padded A row stride (floats)
  constexpr int ABUF = 128 * ALD;     // 2560
  constexpr int BPR  = 258;           // per K-pair stride: 2*128 cols + pad
  constexpr int BUFSZ = ABUF + 8 * BPR;  // 4624
  __shared__ __align__(16) float smem[2 * BUFSZ];

  const int tid  = threadIdx.x;
  const int lane = tid & 31;
  const int wave = tid >> 5;
  const int wm = (wave & 3) * 32;
  const int wn = (wave >> 2) * 64;
  const int m0 = blockIdx.y * 128;
  const int n0 = blockIdx.x * 128;

  const v8f vz = {0.f, 0.f, 0.f, 0.f, 0.f, 0.f, 0.f, 0.f};
  v8f acc[2][4];
#pragma unroll
  for (int i = 0; i < 2; ++i)
#pragma unroll
    for (int j = 0; j < 4; ++j) acc[i][j] = vz;

  const int la_m = tid >> 1;           // 0..127
  const int la_k = (tid & 1) * 8;      // 0 or 8
  const int lb_n = tid & 127;          // 0..127
  const int lb_k = tid >> 7;           // 0 or 1
  const unsigned sbase = lds_off(smem);

  auto issue = [&](int buf, int kt) {
    const float* ap = A + (size_t)(m0 + la_m) * K + kt + la_k;
    const unsigned aoff =
        sbase + (unsigned)(buf * BUFSZ + la_m * ALD + la_k) * 4u;
    async_ld_b128(aoff, ap);            // floats 0..3
    async_ld_b128(aoff + 16u, ap + 4);  // floats 4..7
    const unsigned boff =
        sbase + (unsigned)(buf * BUFSZ + ABUF + 2 * lb_n + lb_k) * 4u;
#pragma unroll
    for (int j = 0; j < 8; ++j) {
      const int kk = lb_k + j * 2;
      async_ld_b32(boff + (unsigned)(j * BPR) * 4u,
                   Bm + (size_t)(kt + kk) * N + n0 + lb_n);
    }
  };

  const int niter = K >> 4;
  issue(0, 0);
  for (int it = 0; it < niter; ++it) {
    const int buf = it & 1;
    const float* as = smem + buf * BUFSZ;
    const float* bs = smem + buf * BUFSZ + ABUF;
    if (it + 1 < niter) {
      issue(buf ^ 1, (it + 1) << 4);
      S_WAIT_ASYNC(0xa);  // oldest stage (10 ops) complete
    } else {
      S_WAIT_ASYNC(0x0);
    }
    __syncthreads();
#pragma unroll
    for (int kc = 0; kc < 4; ++kc) {
      const int kk = kc * 4 + ((lane >> 4) << 1);  // even always
      const int p  = kk >> 1;
      v2f a[2], bf[4];
#pragma unroll
      for (int mi = 0; mi < 2; ++mi)
        a[mi] = ld2(as + (wm + mi * 16 + (lane & 15)) * ALD + kk);
#pragma unroll
      for (int ni = 0; ni < 4; ++ni)
        bf[ni] = ld2(bs + p * BPR + 2 * (wn + ni * 16 + (lane & 15)));
#pragma unroll
      for (int mi = 0; mi < 2; ++mi)
#pragma unroll
        for (int ni = 0; ni < 4; ++ni)
          acc[mi][ni] = wmma4(a[mi], bf[ni], acc[mi][ni]);
    }
    __syncthreads();  // protect buf before it is overwritten next iteration
  }

#pragma unroll
  for (int mi = 0; mi < 2; ++mi)
#pragma unroll
    for (int ni = 0; ni < 4; ++ni) {
      const int colg = n0 + wn + ni * 16 + (lane & 15);
      const float bv = bias ? bias[colg] : 0.0f;
#pragma unroll
      for (int r = 0; r < 8; ++r) {
        const int rowg = m0 + wm + mi * 16 + r + ((lane >> 4) << 3);
        C[(size_t)rowg * N + colg] = acc[mi][ni][r] + bv;
      }
    }
}

// ---------------------------------------------------------------------------
// scores[bh, q, k] = scale * Q·K^T  (causal masked).  128x128 tile per block.
// Only 4 K-iterations (D=64): synchronous staging; pair layouts for mov-free
// b64 fragment reads.
// ---------------------------------------------------------------------------
__global__ __launch_bounds__(256, 2) void attn_scores_128x128(
    const float* __restrict__ qkv, float* __restrict__ probs) {
  constexpr int QLD = 20;
  constexpr int KPR = 258;
  const int tid = threadIdx.x;
  const int bh  = blockIdx.z;
  const int b   = bh >> 4;
  const int h   = bh & 15;
  const int q0  = blockIdx.y * 128;
  const int k0  = blockIdx.x * 128;
  float* out = probs + ((size_t)bh * S_ + q0) * S_ + k0;

  if (k0 > q0 + 127) {  // entirely above causal diagonal
    for (int i = tid; i < 128 * 128; i += 256)
      out[(size_t)(i >> 7) * S_ + (i & 127)] = NEGM_;
    return;
  }

  __shared__ __align__(16) float Qs[128 * QLD];
  __shared__ __align__(16) float Ks[8 * KPR];
  const int lane = tid & 31;
  const int wave = tid >> 5;
  const int wm = (wave & 3) * 32;
  const int wn = (wave >> 2) * 64;

  const float* qb = qkv + (size_t)b * S_ * QKV_LD + h * D_;  // Q base
  const float* kb = qb + E_;                                 // K base

  const v8f vz = {0.f, 0.f, 0.f, 0.f, 0.f, 0.f, 0.f, 0.f};
  v8f acc[2][4];
#pragma unroll
  for (int i = 0; i < 2; ++i)
#pragma unroll
    for (int j = 0; j < 4; ++j) acc[i][j] = vz;

  const int lq_m = tid >> 1;
  const int lq_k = (tid & 1) * 8;
  const int lk_d = tid & 15;       // K-dim index 0..15
  const int lk_c = tid >> 4;       // col base 0..15

  for (int dt = 0; dt < D_; dt += 16) {
    const float* qp = qb + (size_t)(q0 + lq_m) * QKV_LD + dt + lq_k;
#pragma unroll
    for (int j = 0; j < 8; ++j) Qs[lq_m * QLD + lq_k + j] = qp[j];
#pragma unroll
    for (int j = 0; j < 8; ++j) {
      const int col = lk_c + j * 16;
      Ks[(lk_d >> 1) * KPR + 2 * col + (lk_d & 1)] =
          kb[(size_t)(k0 + col) * QKV_LD + dt + lk_d];
    }
    __syncthreads();
#pragma unroll
    for (int kc = 0; kc < 4; ++kc) {
      const int kk = kc * 4 + ((lane >> 4) << 1);
      const int p  = kk >> 1;
      v2f a[2], bf[4];
#pragma unroll
      for (int mi = 0; mi < 2; ++mi)
        a[mi] = ld2(Qs + (wm + mi * 16 + (lane & 15)) * QLD + kk);
#pragma unroll
      for (int ni = 0; ni < 4; ++ni)
        bf[ni] = ld2(Ks + p * KPR + 2 * (wn + ni * 16 + (lane & 15)));
#pragma unroll
      for (int mi = 0; mi < 2; ++mi)
#pragma unroll
        for (int ni = 0; ni < 4; ++ni)
          acc[mi][ni] = wmma4(a[mi], bf[ni], acc[mi][ni]);
    }
    __syncthreads();
  }

#pragma unroll
  for (int mi = 0; mi < 2; ++mi)
#pragma unroll
    for (int ni = 0; ni < 4; ++ni) {
      const int col = wn + ni * 16 + (lane & 15);
#pragma unroll
      for (int r = 0; r < 8; ++r) {
        const int row = wm + mi * 16 + r + ((lane >> 4) << 3);
        const bool keep = (k0 + col) <= (q0 + row);
        out[(size_t)row * S_ + col] = keep ? acc[mi][ni][r] * SCALE_ : NEGM_;
      }
    }
}

// ---------------------------------------------------------------------------
// In-place row softmax over 1024 elems; one wave32 per row, 8 rows/block.
// ---------------------------------------------------------------------------
__global__ __launch_bounds__(256) void softmax_rows(float* __restrict__ probs) {
  const int lane = threadIdx.x & 31;
  const int wave = threadIdx.x >> 5;
  const size_t row = (size_t)blockIdx.x * 8 + wave;
  float* p = probs + row * S_;
  float v[32];
  float mx = -3.4e38f;
#pragma unroll
  for (int i = 0; i < 32; ++i) {
    v[i] = p[lane + i * 32];
    mx = fmaxf(mx, v[i]);
  }
#pragma unroll
  for (int off = 16; off > 0; off >>= 1) mx = fmaxf(mx, __shfl_xor(mx, off, 32));
  float sum = 0.f;
#pragma unroll
  for (int i = 0; i < 32; ++i) {
    v[i] = __expf(v[i] - mx);
    sum += v[i];
  }
#pragma unroll
  for (int off = 16; off > 0; off >>= 1) sum += __shfl_xor(sum, off, 32);
  const float inv = 1.0f / sum;
#pragma unroll
  for (int i = 0; i < 32; ++i) p[lane + i * 32] = v[i] * inv;
}

// ---------------------------------------------------------------------------
// ctx[b, q, h*64+d] = P[bh, q, :] @ V[bh, :, d].  Block tile 128(q) x 64(d),
// 8 waves arranged 4x2 (each wave 32x32 = 2x2 WMMA tiles). K-loop clipped at
// the causal diagonal (P==0 beyond it). Async double-buffered staging:
// per stage per thread 2 async b128 (P) + 4 async b32 (V) = 6 outstanding.
// ---------------------------------------------------------------------------
__global__ __launch_bounds__(256, 2) void attn_ctx_128x64(
    const float* __restrict__ probs, const float* __restrict__ qkv,
    float* __restrict__ ctx) {
  constexpr int PLD = 20;
  constexpr int PBUF = 128 * PLD;       // 2560
  constexpr int VPR = 130;              // per K-pair stride: 2*64 cols + pad
  constexpr int BUFSZ = PBUF + 8 * VPR; // 3600
  __shared__ __align__(16) float smem[2 * BUFSZ];

  const int tid = threadIdx.x;
  const int bh  = blockIdx.z;
  const int b   = bh >> 4;
  const int h   = bh & 15;
  const int q0  = blockIdx.x * 128;

  const int lane = tid & 31;
  const int wave = tid >> 5;
  const int wm = (wave & 3) * 32;
  const int wn = (wave >> 2) * 32;

  const float* pb = probs + ((size_t)bh * S_ + q0) * S_;
  const float* vb = qkv + (size_t)b * S_ * QKV_LD + 2 * E_ + h * D_;

  const v8f vz = {0.f, 0.f, 0.f, 0.f, 0.f, 0.f, 0.f, 0.f};
  v8f acc[2][2];
#pragma unroll
  for (int i = 0; i < 2; ++i)
#pragma unroll
    for (int j = 0; j < 2; ++j) acc[i][j] = vz;

  const int lp_m = tid >> 1;
  const int lp_k = (tid & 1) * 8;
  const int lv_d = tid & 63;
  const int lv_k = tid >> 6;
  const unsigned sbase = lds_off(smem);

  auto issue = [&](int buf, int kt) {
    const float* pp = pb + (size_t)lp_m * S_ + kt + lp_k;
    const unsigned poff =
        sbase + (unsigned)(buf * BUFSZ + lp_m * PLD + lp_k) * 4u;
    async_ld_b128(poff, pp);
    async_ld_b128(poff + 16u, pp + 4);
#pragma unroll
    for (int j = 0; j < 4; ++j) {
      const int kk = lv_k + j * 4;
      const unsigned voff =
          sbase + (unsigned)(buf * BUFSZ + PBUF + (kk >> 1) * VPR +
                             2 * lv_d + (kk & 1)) * 4u;
      async_ld_b32(voff, vb + (size_t)(kt + kk) * QKV_LD + lv_d);
    }
  };

  const int niter = (q0 + 128) >> 4;  // causal clip
  issue(0, 0);
  for (int it = 0; it < niter; ++it) {
    const int buf = it & 1;
    const float* ps = smem + buf * BUFSZ;
    const float* vs = smem + buf * BUFSZ + PBUF;
    if (it + 1 < niter) {
      issue(buf ^ 1, (it + 1) << 4);
      S_WAIT_ASYNC(0x6);  // oldest stage (6 ops) complete
    } else {
      S_WAIT_ASYNC(0x0);
    }
    __syncthreads();
#pragma unroll
    for (int kc = 0; kc < 4; ++kc) {
      const int kk = kc * 4 + ((lane >> 4) << 1);
      const int p  = kk >> 1;
      v2f a[2], bf[2];
#pragma unroll
      for (int mi = 0; mi < 2; ++mi)
        a[mi] = ld2(ps + (wm + mi * 16 + (lane & 15)) * PLD + kk);
#pragma unroll
      for (int ni = 0; ni < 2; ++ni)
        bf[ni] = ld2(vs + p * VPR + 2 * (wn + ni * 16 + (lane & 15)));
#pragma unroll
      for (int mi = 0; mi < 2; ++mi)
#pragma unroll
        for (int ni = 0; ni < 2; ++ni)
          acc[mi][ni] = wmma4(a[mi], bf[ni], acc[mi][ni]);
    }
    __syncthreads();
  }

#pragma unroll
  for (int mi = 0; mi < 2; ++mi)
#pragma unroll
    for (int ni = 0; ni < 2; ++ni) {
      const int col = wn + ni * 16 + (lane & 15);
#pragma unroll
      for (int r = 0; r < 8; ++r) {
        const int row = q0 + wm + mi * 16 + r + ((lane >> 4) << 3);
        ctx[((size_t)b * S_ + row) * E_ + h * D_ + col] = acc[mi][ni][r];
      }
    }
}

// ---------------------------------------------------------------------------
extern "C" void kernel_launch(void* const* d_in, const int* in_sizes, int n_in,
                              void* d_out, int out_size, void* d_ws, size_t ws_size,
                              hipStream_t stream) {
  (void)in_sizes; (void)n_in; (void)out_size; (void)ws_size;
  const float* hidden = (const float*)d_in[0];
  const float* W_attn = (const float*)d_in[1];
  const float* b_attn = (const float*)d_in[2];
  const float* W_proj = (const float*)d_in[3];
  const float* b_proj = (const float*)d_in[4];

  float* attn_out = (float*)d_out;                         // [2,1024,1024]
  float* probs    = attn_out + (size_t)B_ * S_ * E_;       // [2,16,1024,1024]
  float* qkv = (float*)d_ws;                               // [2048,3072] ~25.2MB
  float* ctx = qkv + (size_t)B_ * S_ * QKV_LD;             // [2048,1024] ~8.4MB

  // 1) QKV = hidden @ W_attn + b_attn
  gemm_bias_128x128<<<dim3(QKV_LD / 128, (B_ * S_) / 128), 256, 0, stream>>>(
      hidden, W_attn, b_attn, qkv, B_ * S_, QKV_LD, E_);
  // 2) scores (scaled, causal-masked) -> probs buffer
  attn_scores_128x128<<<dim3(S_ / 128, S_ / 128, B_ * H_), 256, 0, stream>>>(
      qkv, probs);
  // 3) softmax rows in place
  softmax_rows<<<(B_ * H_ * S_) / 8, 256, 0, stream>>>(probs);
  // 4) ctx = P @ V
  attn_ctx_128x64<<<dim3(S_ / 128, 1, B_ * H_), 256, 0, stream>>>(
      probs, qkv, ctx);
  // 5) out = ctx @ W_proj + b_proj
  gemm_bias_128x128<<<dim3(E_ / 128, (B_ * S_) / 128), 256, 0, stream>>>(
      ctx, W_proj, b_proj, attn_out, B_ * S_, E_, E_);
}